// GraphSAGENet_53334903882346
// MI455X (gfx1250) — compile-verified
//
#include <hip/hip_runtime.h>
#include <hip/hip_bf16.h>

// ---------------------------------------------------------------------------
// GCN (2-layer) for MI455X / gfx1250.
//   x  [N,256] f32 ; edge_index [2,E] i64 ; W1 [256,128] ; b1[128] ;
//   W2 [128,64] ; b2[64] ; out [N,64] f32
// Pipeline:
//   deg -> dinv -> h1 = x@W1 (WMMA f32) -> scatter-add norm msgs -> +b1,relu
//   -> h2 = x2@W2 (WMMA f32) -> scatter-add -> +b2
// ---------------------------------------------------------------------------

typedef float v2f __attribute__((ext_vector_type(2)));
typedef float v4f __attribute__((ext_vector_type(4)));
typedef float v8f __attribute__((ext_vector_type(8)));

// ---------------------------- degree / dinv --------------------------------

__global__ void gcn_degree_kernel(const long long* __restrict__ dst, int E,
                                  float* __restrict__ deg) {
    int i = blockIdx.x * blockDim.x + threadIdx.x;
    if (i < E) {
        atomicAdd(&deg[(int)dst[i]], 1.0f);
    }
}

__global__ void gcn_dinv_kernel(float* __restrict__ deg, int n) {
    int i = blockIdx.x * blockDim.x + threadIdx.x;
    if (i < n) {
        // +1 accounts for the self-loop appended in the reference.
        deg[i] = rsqrtf(deg[i] + 1.0f);
    }
}

// ------------------------------- WMMA GEMM ---------------------------------
// H[M,N] = A[M,K] @ W[K,N], all f32 row-major, via V_WMMA_F32_16X16X4_F32.
// Block = 256 threads = 8 waves. Wave w owns rows [ (bx*8+w)*16, +16 ),
// all N columns (N/16 accumulator tiles). W is staged through LDS in
// KC=64-row chunks with padded stride (N+16) to split the two half-wave
// bank groups (64-bank LDS).

template <int K, int N>
__global__ __launch_bounds__(256)
void gcn_gemm_wmma_f32(const float* __restrict__ A,
                       const float* __restrict__ W,
                       float* __restrict__ H, int M) {
    constexpr int NT = N / 16;    // output tiles per wave
    constexpr int KC = 64;        // K-chunk staged in LDS
    constexpr int NP = N + 16;    // padded LDS row stride (dwords)
    __shared__ float sW[KC * NP];

    const int tid   = threadIdx.x;
    const int lane  = tid & 31;
    const int wave  = tid >> 5;                 // 0..7
    const int strip = blockIdx.x * 8 + wave;    // 16-row strip id
    const bool active = (strip * 16) < M;

    const int m   = lane & 15;            // A row within tile / B,C col
    const int kb  = (lane >> 4) << 1;     // K sub-offset: 0 (lo half) / 2 (hi)

    v8f acc[NT];
#pragma unroll
    for (int nt = 0; nt < NT; ++nt) acc[nt] = 0.0f;

    const float* Arow = A + (size_t)(strip * 16 + m) * K;

    for (int kc = 0; kc < K; kc += KC) {
        __syncthreads();  // protect sW from previous chunk's readers
        // Cooperative stage of W[kc .. kc+KC) into padded LDS.
        for (int i = tid; i < KC * N; i += 256) {
            int r = i / N;
            int c = i - r * N;
            sW[r * NP + c] = W[(size_t)(kc + r) * N + c];
        }
        __syncthreads();

        if (active) {
#pragma unroll
            for (int k = 0; k < KC; k += 4) {
                // A fragment: lane holds X[row, kc+k+kb .. +1]  (8B aligned)
                v2f a = *(const v2f*)(Arow + kc + k + kb);
#pragma unroll
                for (int nt = 0; nt < NT; ++nt) {
                    v2f b;
                    b.x = sW[(k + kb) * NP + nt * 16 + m];
                    b.y = sW[(k + kb + 1) * NP + nt * 16 + m];
                    acc[nt] = __builtin_amdgcn_wmma_f32_16x16x4_f32(
                        false, a, false, b, (short)0, acc[nt], false, false);
                }
            }
        }
    }

    if (active) {
        // C/D layout: VGPR r -> (M = r + 8*(lane>=16), N = lane&15)
        const int rbase = (lane >> 4) * 8;
        const int col   = lane & 15;
#pragma unroll
        for (int nt = 0; nt < NT; ++nt) {
#pragma unroll
            for (int r = 0; r < 8; ++r) {
                H[(size_t)(strip * 16 + rbase + r) * N + nt * 16 + col] =
                    acc[nt][r];
            }
        }
    }
}

// ------------------------- message gather/scatter --------------------------
// One group of C/4 lanes per edge; lane q handles columns 4q..4q+3.
// Edges e in [0,E) are real edges; e in [E, E+n) are self-loops.

template <int C>
__global__ void gcn_scatter_kernel(const float* __restrict__ h,
                                   const long long* __restrict__ src,
                                   const long long* __restrict__ dst,
                                   const float* __restrict__ dinv,
                                   float* __restrict__ agg,
                                   int E, int n) {
    constexpr int TPE = C / 4;
    long long t = (long long)blockIdx.x * blockDim.x + threadIdx.x;
    long long e = t / TPE;
    int q = (int)(t - e * TPE);
    if (e >= (long long)E + n) return;

    int s, d;
    float norm;
    if (e < E) {
        s = (int)src[e];
        d = (int)dst[e];
        norm = dinv[s] * dinv[d];
    } else {
        s = d = (int)(e - E);
        float di = dinv[s];
        norm = di * di;
    }

    v4f hv = *(const v4f*)(h + (size_t)s * C + q * 4);
    float* out = agg + (size_t)d * C + q * 4;
    atomicAdd(out + 0, hv.x * norm);
    atomicAdd(out + 1, hv.y * norm);
    atomicAdd(out + 2, hv.z * norm);
    atomicAdd(out + 3, hv.w * norm);
}

// ----------------------------- bias (+ relu) -------------------------------

template <int C, bool RELU>
__global__ void gcn_bias_kernel(float* __restrict__ out,
                                const float* __restrict__ in,
                                const float* __restrict__ bias, int n) {
    long long i = (long long)blockIdx.x * blockDim.x + threadIdx.x;
    if (i >= (long long)n * C) return;
    int c = (int)(i % C);
    float v = in[i] + bias[c];
    if (RELU) v = fmaxf(v, 0.0f);
    out[i] = v;
}

// ------------------------------ launch glue --------------------------------

extern "C" void kernel_launch(void* const* d_in, const int* in_sizes, int n_in,
                              void* d_out, int out_size, void* d_ws, size_t ws_size,
                              hipStream_t stream) {
    (void)n_in; (void)out_size; (void)ws_size;

    const float*     x    = (const float*)d_in[0];       // [N,256]
    const long long* ei   = (const long long*)d_in[1];   // [2,E] int64
    const float*     W1   = (const float*)d_in[2];       // [256,128]
    const float*     b1   = (const float*)d_in[3];       // [128]
    const float*     W2   = (const float*)d_in[4];       // [128,64]
    const float*     b2   = (const float*)d_in[5];       // [64]
    float*           out  = (float*)d_out;               // [N,64]

    const int IN_C = 256, HID = 128, OUT_C = 64;
    const int N = in_sizes[0] / IN_C;        // 50000
    const int E = in_sizes[1] / 2;           // 800000
    const long long* src = ei;
    const long long* dst = ei + E;

    // Workspace layout (floats): deg[N] | h1[N*128] (reused as h2[N*64]) | agg1[N*128]
    float* deg  = (float*)d_ws;
    float* h1   = deg + N;
    float* agg1 = h1 + (size_t)N * HID;
    float* h2   = h1;  // h1 dead after layer-1 scatter; reuse for h2

    hipMemsetAsync(deg,  0, (size_t)N * sizeof(float), stream);
    hipMemsetAsync(agg1, 0, (size_t)N * HID * sizeof(float), stream);
    hipMemsetAsync(out,  0, (size_t)N * OUT_C * sizeof(float), stream);

    // degrees + symmetric norm factors
    gcn_degree_kernel<<<(E + 255) / 256, 256, 0, stream>>>(dst, E, deg);
    gcn_dinv_kernel<<<(N + 255) / 256, 256, 0, stream>>>(deg, N);

    // layer 1: h1 = x @ W1
    {
        int strips = (N + 15) / 16;
        int blocks = (strips + 7) / 8;
        gcn_gemm_wmma_f32<256, 128><<<blocks, 256, 0, stream>>>(x, W1, h1, N);
    }
    // scatter normalized messages into agg1
    {
        long long work = ((long long)E + N) * (HID / 4);
        int blocks = (int)((work + 255) / 256);
        gcn_scatter_kernel<128><<<blocks, 256, 0, stream>>>(h1, src, dst, deg,
                                                            agg1, E, N);
    }
    // x2 = relu(agg1 + b1), in place
    {
        long long work = (long long)N * HID;
        int blocks = (int)((work + 255) / 256);
        gcn_bias_kernel<128, true><<<blocks, 256, 0, stream>>>(agg1, agg1, b1, N);
    }
    // layer 2: h2 = x2 @ W2
    {
        int strips = (N + 15) / 16;
        int blocks = (strips + 7) / 8;
        gcn_gemm_wmma_f32<128, 64><<<blocks, 256, 0, stream>>>(agg1, W2, h2, N);
    }
    // scatter into out (zeroed above)
    {
        long long work = ((long long)E + N) * (OUT_C / 4);
        int blocks = (int)((work + 255) / 256);
        gcn_scatter_kernel<64><<<blocks, 256, 0, stream>>>(h2, src, dst, deg,
                                                           out, E, N);
    }
    // out += b2 (no relu), in place
    {
        long long work = (long long)N * OUT_C;
        int blocks = (int)((work + 255) / 256);
        gcn_bias_kernel<64, false><<<blocks, 256, 0, stream>>>(out, out, b2, N);
    }
}